// DetectionLoss_18743237280404
// MI455X (gfx1250) — compile-verified
//
#include <hip/hip_runtime.h>
#include <math.h>

typedef __attribute__((ext_vector_type(2))) float v2f;
typedef __attribute__((ext_vector_type(8))) float v8f;

// ---------------- problem constants ----------------
#define K_NC    80      // num classes
#define K_NCH   85      // 5 + nc
#define K_B     16      // batch
#define K_NA    3       // anchors per scale
#define K_NT    300     // targets
#define K_CELLS 403200  // 16*3*(80^2+40^2+20^2)
#define K_MASKW (K_CELLS / 32)
// float accumulators after the mask (32-bit units):
// acc[0]=obj weighted sum, acc[1..3]=box_sum[s], acc[4..6]=cls_sum[s], acc[7..9]=nkeep[s]
#define K_NACC  10

__constant__ float c_anc[3][3][2] = {
    {{10.f, 13.f}, {16.f, 30.f}, {33.f, 23.f}},
    {{30.f, 61.f}, {62.f, 45.f}, {59.f, 119.f}},
    {{116.f, 90.f}, {156.f, 198.f}, {373.f, 326.f}}};
__constant__ float c_stride[3] = {8.f, 16.f, 32.f};
__constant__ int   c_grid[3]   = {80, 40, 20};
__constant__ int   c_base[3]   = {0, 307200, 384000};
__constant__ float c_invN[3]   = {1.f / 307200.f, 1.f / 76800.f, 1.f / 19200.f};

// ---------------- helpers ----------------
__device__ __forceinline__ float softplus_f(float x) {
    return fmaxf(x, 0.f) + log1pf(expf(-fabsf(x)));
}
__device__ __forceinline__ float sigmoid_f(float x) {
    return 1.f / (1.f + expf(-x));
}

// Exact f32 wave32 sum using the CDNA5 matrix core:
// A(16x4): lane m vgpr0 -> A[m][0], lane 16+m vgpr0 -> A[m][2]; vgpr1 = 0.
// B(4x16) = ones  ==>  D[m][n] = x_m + x_{16+m}  (same for every column n).
// Sum of D column 0 = lanes 0 and 16, 8 accumulator VGPRs each = full wave sum.
__device__ __forceinline__ float wave_reduce_f32(float x) {
    v2f a; a[0] = x;   a[1] = 0.f;
    v2f b; b[0] = 1.f; b[1] = 1.f;
    v8f c = {0.f, 0.f, 0.f, 0.f, 0.f, 0.f, 0.f, 0.f};
    c = __builtin_amdgcn_wmma_f32_16x16x4_f32(false, a, false, b, (short)0, c,
                                              false, false);
    float s = ((c[0] + c[1]) + (c[2] + c[3])) + ((c[4] + c[5]) + (c[6] + c[7]));
    return __shfl(s, 0, 32) + __shfl(s, 16, 32);
}

__device__ __forceinline__ bool keep_target(int s, int a, float w, float h,
                                            float img) {
    float rw = (w * img) / c_anc[s][a][0];
    float rh = (h * img) / c_anc[s][a][1];
    float mw = fmaxf(rw, 1.f / fmaxf(rw, 1e-8f));
    float mh = fmaxf(rh, 1.f / fmaxf(rh, 1e-8f));
    return fmaxf(mw, mh) < 4.0f;
}

// ---------------- kernels ----------------
__global__ void dl_init(unsigned int* ws) {
    int i = blockIdx.x * blockDim.x + threadIdx.x;
    if (i < K_MASKW + K_NACC) ws[i] = 0u;
}

// One thread per (scale, anchor, target): keep flag -> nkeep, t_obj bitmask.
__global__ void dl_targets(const float* __restrict__ targets,
                           const int* __restrict__ img_sz,
                           unsigned int* __restrict__ mask,
                           float* __restrict__ acc) {
    int idx = blockIdx.x * blockDim.x + threadIdx.x;
    if (idx >= 3 * K_NA * K_NT) return;
    int s = idx / (K_NA * K_NT);
    int rem = idx % (K_NA * K_NT);
    int a = rem / K_NT;
    int n = rem % K_NT;
    const float* t = targets + n * 6;
    float img = (float)img_sz[0];
    if (!keep_target(s, a, t[4], t[5], img)) return;
    atomicAdd(&acc[7 + s], 1.0f);
    int G  = c_grid[s];
    int bi = (int)t[0];
    int gi = min(max((int)(t[2] * (float)G), 0), G - 1);
    int gj = min(max((int)(t[3] * (float)G), 0), G - 1);
    int cell = c_base[s] + ((bi * K_NA + a) * G + gj) * G + gi;
    atomicOr(&mask[cell >> 5], 1u << (cell & 31));
}

// One thread per objectness cell across all 3 scales; WMMA wave reduction.
__global__ void dl_obj(const float* __restrict__ p0,
                       const float* __restrict__ p1,
                       const float* __restrict__ p2,
                       const unsigned int* __restrict__ mask,
                       float* __restrict__ acc) {
    int tid = blockIdx.x * blockDim.x + threadIdx.x;
    float partial = 0.f;
    if (tid < K_CELLS) {
        int s = (tid >= c_base[2]) ? 2 : ((tid >= c_base[1]) ? 1 : 0);
        int local = tid - c_base[s];
        int G = c_grid[s];
        int gyx = G * G;
        int b  = local / (K_NA * gyx);
        int r2 = local % (K_NA * gyx);
        int a  = r2 / gyx;
        int r3 = r2 % gyx;
        const float* pr = (s == 0) ? p0 : ((s == 1) ? p1 : p2);
        float x = pr[(b * (K_NA * K_NCH) + a * K_NCH) * gyx + r3];
        float tv = (float)((mask[tid >> 5] >> (tid & 31)) & 1u);
        partial = (softplus_f(x) - tv * x) * c_invN[s];
    }
    float wsum = wave_reduce_f32(partial);
    if ((threadIdx.x & 31) == 0) atomicAdd(&acc[0], wsum);
}

// One wave32 per (scale, anchor, target) pair: CIoU box loss + class BCE.
__global__ void dl_pairs(const float* __restrict__ p0,
                         const float* __restrict__ p1,
                         const float* __restrict__ p2,
                         const float* __restrict__ targets,
                         const int* __restrict__ img_sz,
                         float* __restrict__ acc) {
    int gwave = (blockIdx.x * blockDim.x + threadIdx.x) >> 5;
    int lane  = threadIdx.x & 31;
    if (gwave >= 3 * K_NA * K_NT) return;
    int s = gwave / (K_NA * K_NT);
    int rem = gwave % (K_NA * K_NT);
    int a = rem / K_NT;
    int n = rem % K_NT;

    const float* t = targets + n * 6;
    float img = (float)img_sz[0];
    float tw = t[4], th = t[5];
    if (!keep_target(s, a, tw, th, img)) return;  // wave-uniform exit

    int G = c_grid[s];
    int gyx = G * G;
    float stride = c_stride[s];
    int bi = (int)t[0];
    int cls = (int)t[1];
    float cx = t[2] * (float)G;
    float cy = t[3] * (float)G;
    float gw = tw * (float)G;
    float gh = th * (float)G;
    int gi = min(max((int)cx, 0), G - 1);
    int gj = min(max((int)cy, 0), G - 1);
    const float* pr = (s == 0) ? p0 : ((s == 1) ? p1 : p2);
    long long off0 = (long long)(bi * (K_NA * K_NCH) + a * K_NCH) * gyx +
                     gj * G + gi;

    // box channels 1..4: lanes 0..3 fetch, broadcast to all lanes
    float v = 0.f;
    if (lane < 4) v = pr[off0 + (long long)(lane + 1) * gyx];
    float px = __shfl(v, 0, 32);
    float py = __shfl(v, 1, 32);
    float pw = __shfl(v, 2, 32);
    float ph = __shfl(v, 3, 32);

    const float eps = 1e-7f;
    float p_cx = sigmoid_f(px) + (float)gi;
    float p_cy = sigmoid_f(py) + (float)gj;
    float p_bw = expf(fminf(fmaxf(pw, -4.f), 4.f)) * (c_anc[s][a][0] / stride);
    float p_bh = expf(fminf(fmaxf(ph, -4.f), 4.f)) * (c_anc[s][a][1] / stride);
    float pb0 = (p_cx - p_bw * 0.5f) * stride;
    float pb1 = (p_cy - p_bh * 0.5f) * stride;
    float pb2 = (p_cx + p_bw * 0.5f) * stride;
    float pb3 = (p_cy + p_bh * 0.5f) * stride;
    float tb0 = (cx - gw * 0.5f) * stride;
    float tb1 = (cy - gh * 0.5f) * stride;
    float tb2 = (cx + gw * 0.5f) * stride;
    float tb3 = (cy + gh * 0.5f) * stride;

    float w1 = pb2 - pb0, h1 = pb3 - pb1;
    float w2 = tb2 - tb0, h2 = tb3 - tb1;
    float ix = fmaxf(fminf(pb2, tb2) - fmaxf(pb0, tb0), 0.f);
    float iy = fmaxf(fminf(pb3, tb3) - fmaxf(pb1, tb1), 0.f);
    float inter = ix * iy;
    float uni = w1 * h1 + w2 * h2 - inter + eps;
    float iou = inter / uni;
    float cw = fmaxf(pb2, tb2) - fminf(pb0, tb0);
    float ch = fmaxf(pb3, tb3) - fminf(pb1, tb1);
    float c2 = cw * cw + ch * ch + eps;
    float dx = tb0 + tb2 - pb0 - pb2;
    float dy = tb1 + tb3 - pb1 - pb3;
    float rho2 = (dx * dx + dy * dy) * 0.25f;
    float dat = atanf(w2 / (h2 + eps)) - atanf(w1 / (h1 + eps));
    float vv = 0.405284734569351f * dat * dat;  // 4/pi^2
    float alpha = vv / (vv - iou + (1.f + eps));
    float ciou = iou - (rho2 / c2 + vv * alpha);
    float box_loss = 1.f - ciou;

    // class BCE: 80 logits swept by the wave
    float cpart = 0.f;
    #pragma unroll
    for (int k = 0; k < 3; ++k) {
        int c = lane + 32 * k;
        if (c < K_NC) {
            float x = pr[off0 + (long long)(5 + c) * gyx];
            float tt = (c == cls) ? 1.f : 0.f;
            cpart += softplus_f(x) - x * tt;
        }
    }
    float csum = wave_reduce_f32(cpart);
    if (lane == 0) {
        atomicAdd(&acc[1 + s], box_loss);
        atomicAdd(&acc[4 + s], csum);
    }
}

__global__ void dl_final(const float* __restrict__ acc,
                         float* __restrict__ out) {
    float loss = acc[0];  // obj already mean-weighted per scale, lambda_obj=1
    #pragma unroll
    for (int s = 0; s < 3; ++s) {
        float nk = fmaxf(acc[7 + s], 1.0f);
        loss += 0.05f * acc[1 + s] / nk;
        loss += 0.5f * acc[4 + s] / (nk * (float)K_NC);
    }
    out[0] = loss;
}

// ---------------- launch ----------------
extern "C" void kernel_launch(void* const* d_in, const int* in_sizes, int n_in,
                              void* d_out, int out_size, void* d_ws,
                              size_t ws_size, hipStream_t stream) {
    const float* p0 = (const float*)d_in[0];
    const float* p1 = (const float*)d_in[1];
    const float* p2 = (const float*)d_in[2];
    const float* tg = (const float*)d_in[3];
    const int* img  = (const int*)d_in[4];

    unsigned int* mask = (unsigned int*)d_ws;
    float* acc = (float*)d_ws + K_MASKW;
    float* out = (float*)d_out;

    int init_n = K_MASKW + K_NACC;
    dl_init<<<(init_n + 255) / 256, 256, 0, stream>>>((unsigned int*)d_ws);

    dl_targets<<<(3 * K_NA * K_NT + 255) / 256, 256, 0, stream>>>(tg, img,
                                                                  mask, acc);

    dl_obj<<<K_CELLS / 256, 256, 0, stream>>>(p0, p1, p2, mask, acc);

    int waves = 3 * K_NA * K_NT;                 // 2700 pairs, 8 waves/block
    dl_pairs<<<(waves + 7) / 8, 256, 0, stream>>>(p0, p1, p2, tg, img, acc);

    dl_final<<<1, 1, 0, stream>>>(acc, out);
}